// AdaptiveFFTAttention_37641093382539
// MI455X (gfx1250) — compile-verified
//
#include <hip/hip_runtime.h>
#include <math.h>

// Problem constants (match reference)
#define DM   1024      // D_MODEL
#define NSEQ 8192      // sequence length
#define BATCH 2
#define NFREQ 4097     // NSEQ/2 + 1

typedef __attribute__((ext_vector_type(2))) float v2f;
typedef __attribute__((ext_vector_type(8))) float v8f;

struct c2 { float x, y; };

__device__ __forceinline__ c2 cmul(c2 a, c2 b) {
  return c2{ a.x * b.x - a.y * b.y, a.x * b.y + a.y * b.x };
}

// LDS byte offset of a shared-memory pointer: per the ISA flat-aperture
// definition, the low 32 bits of a flat LDS address are the LDS offset.
__device__ __forceinline__ unsigned lds_off(const void* p) {
  return (unsigned)(unsigned long long)p;
}

// Async global->LDS copies (GVS mode: saddr base + u32 vgpr byte offset).
// Tracked by ASYNCcnt; wait with async_wait() before the consuming barrier.
__device__ __forceinline__ void async_copy_b64(unsigned lds, unsigned goff,
                                               const float* base) {
  asm volatile("global_load_async_to_lds_b64 %0, %1, %2"
               :: "v"(lds), "v"(goff), "s"(base) : "memory");
}
__device__ __forceinline__ void async_copy_b32(unsigned lds, unsigned goff,
                                               const float* base) {
  asm volatile("global_load_async_to_lds_b32 %0, %1, %2"
               :: "v"(lds), "v"(goff), "s"(base) : "memory");
}
__device__ __forceinline__ void async_wait() {
  asm volatile("s_wait_asynccnt 0" ::: "memory");
}

// ---------------------------------------------------------------------------
// Twiddle table: tw[t] = (cos, sin)(2*pi*t/8192), t in [0,4096).
// ---------------------------------------------------------------------------
__device__ void fill_twiddles(c2* tw) {
  for (int t = threadIdx.x; t < 4096; t += 256) {
    float sn, cs;
    sincosf(6.28318530717958647692f * ((float)t / 8192.f), &sn, &cs);
    tw[t] = c2{ cs, sn };
  }
}

// ---------------------------------------------------------------------------
// Stockham radix-2 FFT, length 8192, 256 threads per workgroup.
// sign = -1.0f forward, +1.0f inverse (unscaled).
// tw: optional LDS twiddle table (nullptr -> compute with sincosf).
// ---------------------------------------------------------------------------
__device__ c2* wg_fft8192(c2* x, c2* y, float sign, const c2* tw) {
  const int tid = threadIdx.x;
  c2* src = x;
  c2* dst = y;
  int nn = 8192;
  for (int ls = 0; nn > 1; ++ls, nn >>= 1) {
    const int s = 1 << ls;     // stride
    const int m = nn >> 1;
    __syncthreads();
    for (int t = tid; t < 4096; t += 256) {
      const int q = t & (s - 1);
      const int p = t >> ls;                 // p in [0, m)
      c2 w;
      if (tw) {
        const c2 tv = tw[p << ls];           // angle 2*pi*p/nn
        w = c2{ tv.x, sign * tv.y };
      } else {
        const float theta = 6.28318530717958647692f * ((float)p / (float)nn);
        float sn, cs;
        sincosf(theta, &sn, &cs);
        w = c2{ cs, sign * sn };
      }
      const c2 a = src[q + s * p];
      const c2 b = src[q + s * (p + m)];
      const c2 sum = c2{ a.x + b.x, a.y + b.y };
      const c2 dif = c2{ a.x - b.x, a.y - b.y };
      dst[q + s * (2 * p)]     = sum;
      dst[q + s * (2 * p + 1)] = cmul(dif, w);
    }
    c2* t2 = src; src = dst; dst = t2;
  }
  __syncthreads();
  return src;   // after 13 stages this is the 'y' argument
}

// ---------------------------------------------------------------------------
// Kernel 1: fused Q/K/V projections (fp32 WMMA 16x16x4).
//   C[m][d] = sum_c X[m][c] * W[d][c] + bias[d]
// Output in head-transposed layout: Out[(b*1024 + d)*8192 + n], m = b*8192+n.
// Block tile: 128 (M) x 64 (N); 256 threads = 8 waves, each wave a 16x64 strip.
// Double-buffered LDS tiles filled by async global->LDS DMA (overlapped with
// WMMA compute); stride-18 rows -> 8B-aligned b64 chunks, conflict-free
// b64 fragment loads (gcd(18,64)=2, no collision within 32 lanes).
// ---------------------------------------------------------------------------
__global__ void __launch_bounds__(256)
proj_gemm_kernel(const float* __restrict__ X,
                 const float* __restrict__ Wq, const float* __restrict__ bq,
                 const float* __restrict__ Wk, const float* __restrict__ bk,
                 const float* __restrict__ Wv, const float* __restrict__ bv,
                 float* __restrict__ Qt, float* __restrict__ Kt,
                 float* __restrict__ Vt)
{
  const int z = blockIdx.z;
  const float* W    = (z == 0) ? Wq : (z == 1) ? Wk : Wv;
  const float* bias = (z == 0) ? bq : (z == 1) ? bk : bv;
  float*       Out  = (z == 0) ? Qt : (z == 1) ? Kt : Vt;

  __shared__ float As[2][128 * 18];   // [mi][ci], stride 18
  __shared__ float Bs[2][64 * 18];    // [ni][ci], stride 18

  const int tid  = threadIdx.x;
  const int lane = tid & 31;
  const int wv8  = tid >> 5;       // wave id 0..7
  const int hi   = lane >> 4;      // lane half
  const int lm   = lane & 15;
  const int blockM = blockIdx.x * 128;
  const int blockN = blockIdx.y * 64;

  // Per-thread chunk coordinates for tile staging (b64 = 2 floats).
  const int amC = (tid >> 3);          // A: mi for chunk set 0 (each +32/r)
  const int ac2 = (tid & 7) * 2;       // A: ci pair base
  const int bnC = (tid >> 3);
  const int bc2 = (tid & 7) * 2;

  auto issue_tiles = [&](int ib, int k0) {
#pragma unroll
    for (int r = 0; r < 4; ++r) {
      const int mi = amC + 32 * r;                  // 0..127
      async_copy_b64(lds_off(&As[ib][mi * 18 + ac2]),
                     (unsigned)(((blockM + mi) * DM + k0 + ac2) * 4), X);
    }
#pragma unroll
    for (int r = 0; r < 2; ++r) {
      const int ni = bnC + 32 * r;                  // 0..63
      async_copy_b64(lds_off(&Bs[ib][ni * 18 + bc2]),
                     (unsigned)(((blockN + ni) * DM + k0 + bc2) * 4), W);
    }
  };

  v8f acc[4] = {};

  issue_tiles(0, 0);
  async_wait();
  __syncthreads();

  for (int k0 = 0, ib = 0; k0 < DM; k0 += 16, ib ^= 1) {
    if (k0 + 16 < DM) issue_tiles(ib ^ 1, k0 + 16);   // overlap DMA with math
#pragma unroll
    for (int kk = 0; kk < 16; kk += 4) {
      const int ka = kk + 2 * hi;                 // this lane half's K pair
      v2f a;
      a.x = As[ib][(wv8 * 16 + lm) * 18 + ka];
      a.y = As[ib][(wv8 * 16 + lm) * 18 + ka + 1];
#pragma unroll
      for (int ns = 0; ns < 4; ++ns) {
        v2f b;
        b.x = Bs[ib][(ns * 16 + lm) * 18 + ka];
        b.y = Bs[ib][(ns * 16 + lm) * 18 + ka + 1];
        acc[ns] = __builtin_amdgcn_wmma_f32_16x16x4_f32(
            false, a, false, b, (short)0, acc[ns], false, false);
      }
    }
    async_wait();        // next tile resident in LDS
    __syncthreads();     // all waves done with current tile
  }

#pragma unroll
  for (int ns = 0; ns < 4; ++ns) {
    const int d = blockN + ns * 16 + lm;
    const float bv_ = bias[d];
#pragma unroll
    for (int j = 0; j < 8; ++j) {
      const int m  = blockM + wv8 * 16 + hi * 8 + j;
      const int bb = m >> 13;          // batch
      const int n  = m & 8191;         // position
      Out[(size_t)(bb * DM + d) * NSEQ + n] = acc[ns][j] + bv_;
    }
  }
}

// ---------------------------------------------------------------------------
// Kernel 2a: partial cross-spectrum. Grid = 32 (b,h) x 8 parts = 256 WGs.
// Each WG FFTs 8 d_k channels (q + i*k packed) and accumulates
// sum_k Qf*conj(Kf) IN REGISTERS (thread owns bins j == tid mod 256),
// then writes a partial spectrum to global.
// LDS: 2 x 64KB ping-pong + 32KB twiddle table = 160KB.
// ---------------------------------------------------------------------------
__global__ void __launch_bounds__(256)
partial_spectra_kernel(const float* __restrict__ Qt,
                       const float* __restrict__ Kt,
                       c2* __restrict__ Pf)
{
  __shared__ c2 bufA[8192];
  __shared__ c2 bufB[8192];
  __shared__ c2 tw[4096];

  const int tid  = threadIdx.x;
  const int wg   = blockIdx.x;       // 0..255
  const int bh   = wg >> 3;          // 0..31
  const int part = wg & 7;           // 0..7  -> channels part*8 .. part*8+7

  fill_twiddles(tw);

  c2 acc[16];
#pragma unroll
  for (int i = 0; i < 16; ++i) acc[i] = c2{ 0.f, 0.f };
  c2 s_ny = c2{ 0.f, 0.f };

  for (int kk = 0; kk < 8; ++kk) {
    const int k = part * 8 + kk;
    const float* qp = Qt + (size_t)(bh * 64 + k) * NSEQ;
    const float* kp = Kt + (size_t)(bh * 64 + k) * NSEQ;
    for (int n = tid; n < NSEQ; n += 256) bufA[n] = c2{ qp[n], kp[n] };
    c2* Z = wg_fft8192(bufA, bufB, -1.f, tw);      // FFT(q + i*k)
#pragma unroll
    for (int i = 0; i < 16; ++i) {
      const int j  = tid + 256 * i;                // 0..4095
      const int j2 = (NSEQ - j) & (NSEQ - 1);
      const c2 zj = Z[j], zc = Z[j2];
      const c2 qf = c2{ 0.5f * (zj.x + zc.x), 0.5f * (zj.y - zc.y) };
      const c2 dd = c2{ zj.x - zc.x, zj.y + zc.y };
      const c2 kf = c2{ 0.5f * dd.y, -0.5f * dd.x };
      acc[i].x += qf.x * kf.x + qf.y * kf.y;       // Qf * conj(Kf)
      acc[i].y += qf.y * kf.x - qf.x * kf.y;
    }
    if (tid == 0) {                                 // Nyquist bin j = 4096
      const c2 zj = Z[4096];
      s_ny.x += zj.x * zj.y;                        // Qf, Kf real at Nyquist
    }
  }

  c2* out = Pf + (size_t)wg * NFREQ;
#pragma unroll
  for (int i = 0; i < 16; ++i) out[tid + 256 * i] = acc[i];
  if (tid == 0) out[4096] = s_ny;
}

// ---------------------------------------------------------------------------
// Kernel 2b: per (b,h): combine 8 partials -> hermitian spectrum,
// irfft -> scores, softmax -> attn, rfft -> Af.  Grid = 32 (light kernel).
// ---------------------------------------------------------------------------
__global__ void __launch_bounds__(256)
combine_softmax_kernel(const c2* __restrict__ Pf, c2* __restrict__ Af)
{
  __shared__ c2 bufA[8192];
  __shared__ c2 bufB[8192];
  __shared__ float red[256];

  const int tid = threadIdx.x;
  const int bh  = blockIdx.x;        // 0..31

  const c2* base = Pf + (size_t)(bh * 8) * NFREQ;
  for (int j = tid; j < NFREQ; j += 256) {
    c2 s = c2{ 0.f, 0.f };
#pragma unroll
    for (int part = 0; part < 8; ++part) {
      const c2 p = base[(size_t)part * NFREQ + j];
      s.x += p.x; s.y += p.y;
    }
    bufA[j] = s;
    if (j != 0 && j != 4096) bufA[NSEQ - j] = c2{ s.x, -s.y };
  }
  c2* R = wg_fft8192(bufA, bufB, 1.f, nullptr);   // scores = R[n].x / 8192

  const float sscale = (1.f / 8192.f) * 0.125f;   // irfft norm * 1/sqrt(d_k)

  float lmax = -3.4e38f;
  for (int n = tid; n < NSEQ; n += 256) lmax = fmaxf(lmax, R[n].x * sscale);
  red[tid] = lmax;
  __syncthreads();
  for (int off = 128; off > 0; off >>= 1) {
    if (tid < off) red[tid] = fmaxf(red[tid], red[tid + off]);
    __syncthreads();
  }
  const float gmax = red[0];
  __syncthreads();

  float lsum = 0.f;
  for (int n = tid; n < NSEQ; n += 256) {
    const float e = expf(R[n].x * sscale - gmax);
    bufA[n].x = e;
    lsum += e;
  }
  red[tid] = lsum;
  __syncthreads();
  for (int off = 128; off > 0; off >>= 1) {
    if (tid < off) red[tid] += red[tid + off];
    __syncthreads();
  }
  const float rs = 1.f / red[0];

  for (int n = tid; n < NSEQ; n += 256) bufA[n] = c2{ bufA[n].x * rs, 0.f };
  c2* A2 = wg_fft8192(bufA, bufB, -1.f, nullptr); // Af = rfft(attn)
  for (int j = tid; j < NFREQ; j += 256) Af[(size_t)bh * NFREQ + j] = A2[j];
}

// ---------------------------------------------------------------------------
// Kernel 3: per (b,h) and pair of d_k channels: O = irfft(Af * rfft(V)).
// Packs two real channels per complex FFT; grid = 32 * 32 = 1024 workgroups.
// LDS: ping-pong 128KB + 32KB twiddle table = 160KB.
// ---------------------------------------------------------------------------
__global__ void __launch_bounds__(256)
apply_attn_kernel(const float* __restrict__ Vt,
                  const c2* __restrict__ Af,
                  float* __restrict__ Ot)
{
  __shared__ c2 bufA[8192];
  __shared__ c2 bufB[8192];
  __shared__ c2 tw[4096];

  const int tid = threadIdx.x;
  const int w   = blockIdx.x;
  const int bh  = w >> 5;
  const int pr  = w & 31;            // pair index: channels 2*pr, 2*pr+1

  fill_twiddles(tw);

  const float* v0 = Vt + (size_t)(bh * 64 + 2 * pr) * NSEQ;
  const float* v1 = v0 + NSEQ;
  for (int n = tid; n < NSEQ; n += 256) bufA[n] = c2{ v0[n], v1[n] };
  c2* Z = wg_fft8192(bufA, bufB, -1.f, tw);

  const c2* Afp = Af + (size_t)bh * NFREQ;
  for (int j = tid; j < NFREQ; j += 256) {
    const int j2 = (NSEQ - j) & (NSEQ - 1);
    const c2 zj = Z[j], zc = Z[j2];
    const c2 vf0 = c2{ 0.5f * (zj.x + zc.x), 0.5f * (zj.y - zc.y) };
    const c2 dd  = c2{ zj.x - zc.x, zj.y + zc.y };
    const c2 vf1 = c2{ 0.5f * dd.y, -0.5f * dd.x };
    const c2 a   = Afp[j];
    const c2 h0  = cmul(a, vf0);
    const c2 h1  = cmul(a, vf1);
    // G = H0 + i*H1 (both hermitian) -> single inverse FFT yields both reals
    bufA[j] = c2{ h0.x - h1.y, h0.y + h1.x };
    if (j != 0 && j != 4096)
      bufA[NSEQ - j] = c2{ h0.x + h1.y, h1.x - h0.y };
  }
  c2* R = wg_fft8192(bufA, bufB, 1.f, tw);

  const float inv = 1.f / 8192.f;
  float* o0 = Ot + (size_t)(bh * 64 + 2 * pr) * NSEQ;
  float* o1 = o0 + NSEQ;
  for (int n = tid; n < NSEQ; n += 256) {
    o0[n] = R[n].x * inv;
    o1[n] = R[n].y * inv;
  }
}

// ---------------------------------------------------------------------------
// Kernel 4: output GEMM  Y[m][d] = sum_c O[m][c] * Wo[d][c] + bo[d]
// A gathered from head-transposed Ot (contiguous in n) via per-element
// async b32 copies; B staged with async b64; double-buffered.
// ---------------------------------------------------------------------------
__global__ void __launch_bounds__(256)
out_gemm_kernel(const float* __restrict__ Ot,
                const float* __restrict__ Wo,
                const float* __restrict__ bo,
                float* __restrict__ Y)
{
  __shared__ float As[2][128 * 18];
  __shared__ float Bs[2][64 * 18];

  const int tid  = threadIdx.x;
  const int lane = tid & 31;
  const int wv8  = tid >> 5;
  const int hi   = lane >> 4;
  const int lm   = lane & 15;
  const int blockM = blockIdx.x * 128;
  const int blockN = blockIdx.y * 64;
  const int bb = blockM >> 13;        // tile never crosses batch (8192 % 128 == 0)
  const int n0 = blockM & 8191;

  const int aci = tid >> 7;            // 0..1 (A: ci base, +2 per r)
  const int ami = tid & 127;           // A: mi (coalesced n)
  const int bnC = tid >> 3;
  const int bc2 = (tid & 7) * 2;

  auto issue_tiles = [&](int ib, int k0) {
#pragma unroll
    for (int r = 0; r < 8; ++r) {
      const int ci = aci + 2 * r;                   // 0..15
      async_copy_b32(lds_off(&As[ib][ami * 18 + ci]),
                     (unsigned)(((bb * DM + k0 + ci) * NSEQ + n0 + ami) * 4), Ot);
    }
#pragma unroll
    for (int r = 0; r < 2; ++r) {
      const int ni = bnC + 32 * r;
      async_copy_b64(lds_off(&Bs[ib][ni * 18 + bc2]),
                     (unsigned)(((blockN + ni) * DM + k0 + bc2) * 4), Wo);
    }
  };

  v8f acc[4] = {};

  issue_tiles(0, 0);
  async_wait();
  __syncthreads();

  for (int k0 = 0, ib = 0; k0 < DM; k0 += 16, ib ^= 1) {
    if (k0 + 16 < DM) issue_tiles(ib ^ 1, k0 + 16);
#pragma unroll
    for (int kk = 0; kk < 16; kk += 4) {
      const int ka = kk + 2 * hi;
      v2f a;
      a.x = As[ib][(wv8 * 16 + lm) * 18 + ka];
      a.y = As[ib][(wv8 * 16 + lm) * 18 + ka + 1];
#pragma unroll
      for (int ns = 0; ns < 4; ++ns) {
        v2f b;
        b.x = Bs[ib][(ns * 16 + lm) * 18 + ka];
        b.y = Bs[ib][(ns * 16 + lm) * 18 + ka + 1];
        acc[ns] = __builtin_amdgcn_wmma_f32_16x16x4_f32(
            false, a, false, b, (short)0, acc[ns], false, false);
      }
    }
    async_wait();
    __syncthreads();
  }

#pragma unroll
  for (int ns = 0; ns < 4; ++ns) {
    const int d = blockN + ns * 16 + lm;
    const float bv_ = bo[d];
#pragma unroll
    for (int j = 0; j < 8; ++j) {
      const int m = blockM + wv8 * 16 + hi * 8 + j;
      Y[(size_t)m * DM + d] = acc[ns][j] + bv_;
    }
  }
}

// ---------------------------------------------------------------------------
// Host-side launcher.
// Inputs: x, Wq, bq, Wk, bk, Wv, bv, Wo, bo  (all fp32).
// Workspace (floats): Qt[16.7M] Kt[16.7M] Vt[16.7M] Af[262K] Pf[2.1M]
//   (~211 MB total; Ot reuses Qt).
// ---------------------------------------------------------------------------
extern "C" void kernel_launch(void* const* d_in, const int* in_sizes, int n_in,
                              void* d_out, int out_size, void* d_ws, size_t ws_size,
                              hipStream_t stream) {
  (void)in_sizes; (void)n_in; (void)out_size; (void)ws_size;

  const float* x  = (const float*)d_in[0];
  const float* Wq = (const float*)d_in[1];
  const float* bq = (const float*)d_in[2];
  const float* Wk = (const float*)d_in[3];
  const float* bk = (const float*)d_in[4];
  const float* Wv = (const float*)d_in[5];
  const float* bv = (const float*)d_in[6];
  const float* Wo = (const float*)d_in[7];
  const float* bo = (const float*)d_in[8];
  float* Y = (float*)d_out;

  const size_t chElems = (size_t)BATCH * DM * NSEQ;   // 16,777,216 floats
  float* Qt = (float*)d_ws;
  float* Kt = Qt + chElems;
  float* Vt = Kt + chElems;
  c2*    Af = (c2*)(Vt + chElems);                    // 32 * 4097 complex
  c2*    Pf = Af + (size_t)32 * NFREQ;                // 256 * 4097 complex

  dim3 blk(256);
  dim3 g1((BATCH * NSEQ) / 128, DM / 64, 3);
  proj_gemm_kernel<<<g1, blk, 0, stream>>>(x, Wq, bq, Wk, bk, Wv, bv, Qt, Kt, Vt);

  partial_spectra_kernel<<<dim3(256), blk, 0, stream>>>(Qt, Kt, Pf);

  combine_softmax_kernel<<<dim3(32), blk, 0, stream>>>(Pf, Af);

  apply_attn_kernel<<<dim3(1024), blk, 0, stream>>>(Vt, Af, /*Ot=*/Qt);

  dim3 g2((BATCH * NSEQ) / 128, DM / 64, 1);
  out_gemm_kernel<<<g2, blk, 0, stream>>>(/*Ot=*/Qt, Wo, bo, Y);
}